// ChamferDistanceL2_58342835749036
// MI455X (gfx1250) — compile-verified
//
#include <hip/hip_runtime.h>

typedef float v2f __attribute__((ext_vector_type(2)));
typedef float v8f __attribute__((ext_vector_type(8)));

#define B_  4
#define N_  8192
#define M_  8192
#define BIGF 3.0e38f

#define CHUNK_COLS    2048
#define CHUNK_TILES   (CHUNK_COLS / 16)      // 128 tiles per chunk
#define CHUNK_ENTRIES (CHUNK_COLS * 2)       // 4096 float2 = 32 KB per chunk

// ws layout: [0..3] float accumulators {S1, W1, S2, W2} @0;
// pack1 (float4 [B*N]) @256; pack2 (float4 [B*M]);
// bpack1 (float2 [B*N*2], WMMA-B-operand layout); bpack2 (float2 [B*M*2]).

__global__ void zero_accums_kernel(float* acc) {
    if (threadIdx.x < 4) acc[threadIdx.x] = 0.0f;
}

// Pack (x,y,z,||p||^2) float4 per point, plus the B-operand swizzled layout:
// per 16-col tile, 32 float2 entries: lane l<16 -> (x,z) = (K0,K2) of col l;
// lane l>=16 -> (y,sq) = (K1,K3) of col l-16.
__global__ __launch_bounds__(256) void prep_kernel(
    const float* __restrict__ xyz1, const float* __restrict__ xyz2,
    const float* __restrict__ w1,   const float* __restrict__ w2,
    float4* __restrict__ pack1, float4* __restrict__ pack2,
    float2* __restrict__ bpack1, float2* __restrict__ bpack2,
    float* __restrict__ acc)
{
    const int i    = blockIdx.x * blockDim.x + threadIdx.x;
    const int lane = threadIdx.x & 31;
    const int sub  = i & 15;                  // batch stride divisible by 16

    float wa = 0.0f, wb = 0.0f;
    if (i < B_ * N_) {
        float x = xyz1[3*i], y = xyz1[3*i+1], z = xyz1[3*i+2];
        float sq = x*x + y*y + z*z;
        pack1[i] = make_float4(x, y, z, sq);
        const int idx0 = 2 * (i - sub) + sub;
        bpack1[idx0]      = make_float2(x, z);
        bpack1[idx0 + 16] = make_float2(y, sq);
        wa = w1[i];
    }
    if (i < B_ * M_) {
        float x = xyz2[3*i], y = xyz2[3*i+1], z = xyz2[3*i+2];
        float sq = x*x + y*y + z*z;
        pack2[i] = make_float4(x, y, z, sq);
        const int idx0 = 2 * (i - sub) + sub;
        bpack2[idx0]      = make_float2(x, z);
        bpack2[idx0 + 16] = make_float2(y, sq);
        wb = w2[i];
    }
    #pragma unroll
    for (int off = 1; off < 32; off <<= 1) {
        wa += __shfl_xor(wa, off, 32);
        wb += __shfl_xor(wb, off, 32);
    }
    if (lane == 0) {
        atomicAdd(&acc[1], wa);
        atomicAdd(&acc[3], wb);
    }
}

__device__ __forceinline__ void wait_async0() {
#if __has_builtin(__builtin_amdgcn_s_wait_asynccnt)
    __builtin_amdgcn_s_wait_asynccnt(0);
#else
    asm volatile("s_wait_asynccnt 0x0" ::: "memory");
#endif
}

// Stage CHUNK_ENTRIES float2 (32 KB) from global into LDS with async-to-LDS DMA.
// GVS form: vdst = per-lane LDS byte offset, vaddr = per-lane 32-bit byte offset,
// saddr = uniform 64-bit base (SGPR pair).
__device__ __forceinline__ void async_stage(const float2* __restrict__ gsrc,
                                            float2* lds_dst, int tid) {
    unsigned lbase =
        (unsigned)(size_t)(__attribute__((address_space(3))) float2*)lds_dst;
    unsigned long long gbase = (unsigned long long)gsrc;
    #pragma unroll
    for (int k = 0; k < CHUNK_ENTRIES / 256; ++k) {
        unsigned e = (unsigned)(k * 256 + tid);
        asm volatile("global_load_async_to_lds_b64 %0, %1, %2"
                     :: "v"(lbase + e * 8u), "v"(e * 8u), "s"(gbase)
                     : "memory");
    }
}

// One pass: rows of rowPack vs all cols of colB (pre-swizzled B layout).
// Each wave owns TWO 16-row A tiles (32 rows) -> 2 WMMAs per B load.
// Columns stream through a double-buffered LDS chunk pipeline.
__global__ __launch_bounds__(256) void chamfer_pass_kernel(
    const float4* __restrict__ rowPack, const float* __restrict__ wRow,
    const float2* __restrict__ colB, float* __restrict__ accS,
    int rowsPerBatch, int colsPerBatch)
{
    __shared__ float2 lbuf[2][CHUNK_ENTRIES];   // 2 x 32 KB

    const int lane = threadIdx.x & 31;
    const int wave = threadIdx.x >> 5;
    const int blocksPerBatch = rowsPerBatch >> 8;          // 256 rows per block
    const int b       = blockIdx.x / blocksPerBatch;
    const int rowBase = ((blockIdx.x % blocksPerBatch) << 8) + (wave << 5);

    const float4* __restrict__ rpb  = rowPack + (size_t)b * rowsPerBatch;
    const float2* __restrict__ gcol = colB + (size_t)b * colsPerBatch * 2;

    const bool lo  = (lane < 16);
    const int  sub = lane & 15;

    // Two A tiles (rows rowBase..+15 and rowBase+16..+31).
    // A layout (16x4 f32): VGPR0 = {K0 | K2}, VGPR1 = {K1 | K3} across halves;
    // row vector = (-2x, -2y, -2z, 1).
    float4 rp0 = rpb[rowBase + sub];
    float4 rp1 = rpb[rowBase + 16 + sub];
    v2f a0, a1;
    a0.x = lo ? (-2.0f * rp0.x) : (-2.0f * rp0.z);
    a0.y = lo ? (-2.0f * rp0.y) : 1.0f;
    a1.x = lo ? (-2.0f * rp1.x) : (-2.0f * rp1.z);
    a1.y = lo ? (-2.0f * rp1.y) : 1.0f;

    float ma0 = BIGF, ma1 = BIGF, ma2 = BIGF, ma3 = BIGF;
    float ma4 = BIGF, ma5 = BIGF, ma6 = BIGF, ma7 = BIGF;
    float mb0 = BIGF, mb1 = BIGF, mb2 = BIGF, mb3 = BIGF;
    float mb4 = BIGF, mb5 = BIGF, mb6 = BIGF, mb7 = BIGF;

    const int nchunks = colsPerBatch / CHUNK_COLS;

    async_stage(gcol, &lbuf[0][0], threadIdx.x);
    wait_async0();
    __syncthreads();

    for (int ch = 0; ch < nchunks; ++ch) {
        const int cur = ch & 1;
        if (ch + 1 < nchunks)   // safe: other buffer's readers passed last barrier
            async_stage(gcol + (size_t)(ch + 1) * CHUNK_ENTRIES,
                        &lbuf[cur ^ 1][0], threadIdx.x);

        const float2* __restrict__ lb = &lbuf[cur][0];
        #pragma unroll 4
        for (int t = 0; t < CHUNK_TILES; ++t) {
            float2 bvf = lb[t * 32 + lane];
            v2f bv; bv.x = bvf.x; bv.y = bvf.y;

            v8f c = {0.f, 0.f, 0.f, 0.f, 0.f, 0.f, 0.f, 0.f};
            // D[m][n] = sq_col[n] - 2*dot(row m, col n)
            v8f d0 = __builtin_amdgcn_wmma_f32_16x16x4_f32(
                false, a0, false, bv, (short)0, c, false, false);
            v8f d1 = __builtin_amdgcn_wmma_f32_16x16x4_f32(
                false, a1, false, bv, (short)0, c, false, false);

            ma0 = fminf(ma0, d0[0]); ma1 = fminf(ma1, d0[1]);
            ma2 = fminf(ma2, d0[2]); ma3 = fminf(ma3, d0[3]);
            ma4 = fminf(ma4, d0[4]); ma5 = fminf(ma5, d0[5]);
            ma6 = fminf(ma6, d0[6]); ma7 = fminf(ma7, d0[7]);
            mb0 = fminf(mb0, d1[0]); mb1 = fminf(mb1, d1[1]);
            mb2 = fminf(mb2, d1[2]); mb3 = fminf(mb3, d1[3]);
            mb4 = fminf(mb4, d1[4]); mb5 = fminf(mb5, d1[5]);
            mb6 = fminf(mb6, d1[6]); mb7 = fminf(mb7, d1[7]);
        }

        wait_async0();       // my async loads for next chunk are in LDS
        __syncthreads();     // everyone done reading current chunk
    }

    // Cross-lane min within each 16-lane half
    // (rows r in lanes 0-15 hold M=r, lanes 16-31 hold M=r+8 per C/D layout).
    #pragma unroll
    for (int off = 1; off < 16; off <<= 1) {
        ma0 = fminf(ma0, __shfl_xor(ma0, off, 32));
        ma1 = fminf(ma1, __shfl_xor(ma1, off, 32));
        ma2 = fminf(ma2, __shfl_xor(ma2, off, 32));
        ma3 = fminf(ma3, __shfl_xor(ma3, off, 32));
        ma4 = fminf(ma4, __shfl_xor(ma4, off, 32));
        ma5 = fminf(ma5, __shfl_xor(ma5, off, 32));
        ma6 = fminf(ma6, __shfl_xor(ma6, off, 32));
        ma7 = fminf(ma7, __shfl_xor(ma7, off, 32));
        mb0 = fminf(mb0, __shfl_xor(mb0, off, 32));
        mb1 = fminf(mb1, __shfl_xor(mb1, off, 32));
        mb2 = fminf(mb2, __shfl_xor(mb2, off, 32));
        mb3 = fminf(mb3, __shfl_xor(mb3, off, 32));
        mb4 = fminf(mb4, __shfl_xor(mb4, off, 32));
        mb5 = fminf(mb5, __shfl_xor(mb5, off, 32));
        mb6 = fminf(mb6, __shfl_xor(mb6, off, 32));
        mb7 = fminf(mb7, __shfl_xor(mb7, off, 32));
    }

    float minsA[8] = {ma0, ma1, ma2, ma3, ma4, ma5, ma6, ma7};
    float minsB[8] = {mb0, mb1, mb2, mb3, mb4, mb5, mb6, mb7};

    float p = 0.0f;
    if (sub == 0) {   // one leader lane per half: rows +8*half of each tile
        const int h   = lane >> 4;
        const int mA  = rowBase + (h << 3);
        const int mB  = rowBase + 16 + (h << 3);
        const float* wr = wRow + (size_t)b * rowsPerBatch;
        #pragma unroll
        for (int r = 0; r < 8; ++r) {
            float dA = fmaxf(minsA[r] + rpb[mA + r].w, 0.0f);  // + sq_row, clamp
            float dB = fmaxf(minsB[r] + rpb[mB + r].w, 0.0f);
            p += dA * wr[mA + r] + dB * wr[mB + r];
        }
    }
    p += __shfl_xor(p, 16, 32);
    if (lane == 0) atomicAdd(accS, p);
}

__global__ void finalize_kernel(const float* __restrict__ acc, float* __restrict__ out) {
    if (threadIdx.x == 0 && blockIdx.x == 0)
        out[0] = 0.5f * (acc[0] / acc[1] + acc[2] / acc[3]);
}

extern "C" void kernel_launch(void* const* d_in, const int* in_sizes, int n_in,
                              void* d_out, int out_size, void* d_ws, size_t ws_size,
                              hipStream_t stream) {
    const float* xyz1 = (const float*)d_in[0];
    const float* xyz2 = (const float*)d_in[1];
    const float* w1   = (const float*)d_in[2];
    const float* w2   = (const float*)d_in[3];

    float*  acc    = (float*)d_ws;                     // {S1, W1, S2, W2}
    float4* pack1  = (float4*)((char*)d_ws + 256);     // [B*N] float4
    float4* pack2  = pack1 + (size_t)B_ * N_;          // [B*M] float4
    float2* bpack1 = (float2*)(pack2 + (size_t)B_ * M_);   // [B*N*2] float2
    float2* bpack2 = bpack1 + (size_t)B_ * N_ * 2;         // [B*M*2] float2

    zero_accums_kernel<<<1, 32, 0, stream>>>(acc);

    const int npts = (B_ * N_ > B_ * M_) ? B_ * N_ : B_ * M_;
    prep_kernel<<<(npts + 255) / 256, 256, 0, stream>>>(
        xyz1, xyz2, w1, w2, pack1, pack2, bpack1, bpack2, acc);

    // Pass 1: rows = xyz1, cols = xyz2  -> S1 (acc[0])
    chamfer_pass_kernel<<<(B_ * N_) / 256, 256, 0, stream>>>(
        pack1, w1, bpack2, &acc[0], N_, M_);

    // Pass 2: rows = xyz2, cols = xyz1  -> S2 (acc[2])
    chamfer_pass_kernel<<<(B_ * M_) / 256, 256, 0, stream>>>(
        pack2, w2, bpack1, &acc[2], M_, N_);

    finalize_kernel<<<1, 1, 0, stream>>>(acc, (float*)d_out);
}